// MultiBoxLoss_62543313764527
// MI455X (gfx1250) — compile-verified
//
#include <hip/hip_runtime.h>
#include <cstdint>
#include <math.h>

// MultiBoxLoss for MI455X (gfx1250).
// One workgroup (256 threads = 8 wave32) per batch row.
// conf_data streamed via CDNA5 async global->LDS DMA (ASYNCcnt), double-buffered.

#define NUM_P 8732
#define NUM_O 16
#define NUM_C 21
#define TPB   256
#define CHUNK 256
#define NCHUNK       ((NUM_P + CHUNK - 1) / CHUNK)        // 35
#define CHUNK_FLOATS (CHUNK * NUM_C)                      // 5376
#define CHUNK_BYTES  (CHUNK_FLOATS * 4)                   // 21504 (16B multiple)
#define ITERS_FULL   (((CHUNK_BYTES / 16) + TPB - 1) / TPB) // 6

// Low 32 bits of a generic pointer into LDS == LDS byte offset (flat aperture
// maps LDS via addr[31:0]); async-to-LDS VDST VGPR wants exactly that.
__device__ __forceinline__ unsigned ldsOff(const void* p) {
  return (unsigned)(uintptr_t)p;
}

// Issue a uniform number of global_load_async_to_lds_b128 per wave so a single
// s_wait_asynccnt threshold is valid for every wave (in-order completion).
__device__ __forceinline__ void asyncCopyChunk(const float* rowBase, unsigned gByteOff,
                                               const float* lbuf, int n16, int tid) {
  unsigned lbase = ldsOff(lbuf);
  int iters = (n16 + TPB - 1) / TPB;
  for (int it = 0; it < iters; ++it) {
    int i = tid + it * TPB;
    if (i > n16 - 1) i = n16 - 1;          // clamp: duplicate loads are harmless
    unsigned go = gByteOff + (unsigned)i * 16u;
    unsigned lo = lbase + (unsigned)i * 16u;
    asm volatile("global_load_async_to_lds_b128 %0, %1, %2"
                 :: "v"(lo), "v"(go), "s"(rowBase) : "memory");
  }
}

__device__ __forceinline__ float smoothL1(float x) {
  float a = fabsf(x);
  return (a < 1.0f) ? 0.5f * a * a : a - 0.5f;
}

__global__ __launch_bounds__(TPB) void multibox_row_kernel(
    const float* __restrict__ loc, const float* __restrict__ conf,
    const float* __restrict__ priors, const float* __restrict__ truths,
    const int* __restrict__ labels, float* __restrict__ acc) {
  const int tid = threadIdx.x;
  const int b = blockIdx.x;

  extern __shared__ char smem[];
  float*    confbuf  = (float*)smem;                        // 2 * CHUNK_FLOATS
  float*    bto      = confbuf + 2 * CHUNK_FLOATS;          // NUM_P best_truth_overlap
  float*    ceArr    = bto + NUM_P;                         // NUM_P cross-entropy values
  unsigned* ukey     = (unsigned*)(ceArr + NUM_P);          // NUM_P ordered keys
  float*    truths_s = (float*)(ukey + NUM_P);              // NUM_O * 4
  int*      labels_s = (int*)(truths_s + NUM_O * 4);        // NUM_O
  int*      bpi      = labels_s + NUM_O;                    // NUM_O best prior per truth
  unsigned char* bti = (unsigned char*)(bpi + NUM_O);       // NUM_P best_truth_idx

  __shared__ float    red_f[TPB];
  __shared__ int      red_i[TPB];
  __shared__ unsigned hist[256];
  __shared__ int      scan_s[TPB];
  __shared__ int      s_sel, s_krem, s_numpos, s_numneg;

  const float* confRow = conf + (size_t)b * NUM_P * NUM_C;

  if (tid < NUM_O) {
    ((float4*)truths_s)[tid] = ((const float4*)truths)[(size_t)b * NUM_O + tid];
    labels_s[tid] = labels[(size_t)b * NUM_O + tid];
  }

  // Kick off DMA of conf chunk 0 now; it overlaps all of the matching phase.
  asyncCopyChunk(confRow, 0u, confbuf, CHUNK_BYTES / 16, tid);
  __syncthreads();

  // ---------------- Phase A: matching (jaccard / argmax) ----------------
  float tmax[NUM_O];
  int   targ[NUM_O];
#pragma unroll
  for (int o = 0; o < NUM_O; ++o) { tmax[o] = -1.0f; targ[o] = 0; }

  for (int p = tid; p < NUM_P; p += TPB) {
    float4 pr = ((const float4*)priors)[p];       // center form cx,cy,w,h
    float px1 = pr.x - 0.5f * pr.z, py1 = pr.y - 0.5f * pr.w;
    float px2 = pr.x + 0.5f * pr.z, py2 = pr.y + 0.5f * pr.w;
    float areab = pr.z * pr.w;
    float bestov = -1.0f; int besto = 0;
#pragma unroll
    for (int o = 0; o < NUM_O; ++o) {
      float4 t = ((const float4*)truths_s)[o];    // point form
      float ix = fminf(t.z, px2) - fmaxf(t.x, px1); ix = fmaxf(ix, 0.0f);
      float iy = fminf(t.w, py2) - fmaxf(t.y, py1); iy = fmaxf(iy, 0.0f);
      float inter = ix * iy;
      float areaa = (t.z - t.x) * (t.w - t.y);
      float iou = inter / (areaa + areab - inter);
      if (iou > bestov) { bestov = iou; besto = o; }   // first-max tie-break
      if (iou > tmax[o]) { tmax[o] = iou; targ[o] = p; }
    }
    bto[p] = bestov;
    bti[p] = (unsigned char)besto;
  }
  __syncthreads();

  // per-truth argmax over priors (value desc, index asc on ties)
  for (int o = 0; o < NUM_O; ++o) {
    red_f[tid] = tmax[o]; red_i[tid] = targ[o];
    __syncthreads();
    for (int s2 = TPB >> 1; s2 > 0; s2 >>= 1) {
      if (tid < s2) {
        float a = red_f[tid], bv = red_f[tid + s2];
        int ai = red_i[tid], bi2 = red_i[tid + s2];
        if (bv > a || (bv == a && bi2 < ai)) { red_f[tid] = bv; red_i[tid] = bi2; }
      }
      __syncthreads();
    }
    if (tid == 0) bpi[o] = red_i[0];
    __syncthreads();
  }
  if (tid == 0) {
    for (int o = 0; o < NUM_O; ++o) {   // sequential scatter, last write wins
      int pp = bpi[o];
      bto[pp] = 2.0f;
      bti[pp] = (unsigned char)o;
    }
  }
  __syncthreads();

  // ------- Phase B: encode + smooth-L1, LSE/CE, keys (async pipelined) -------
  float myLossL = 0.0f;
  int myNumPos = 0;

  for (int c = 0; c < NCHUNK; ++c) {
    int nxt = c + 1;
    if (nxt < NCHUNK) {
      int np = NUM_P - nxt * CHUNK; if (np > CHUNK) np = CHUNK;
      int n16 = (np * NUM_C * 4) / 16;       // np is a multiple of 4 -> exact
      asyncCopyChunk(confRow, (unsigned)(nxt * CHUNK_BYTES),
                     confbuf + (nxt & 1) * CHUNK_FLOATS, n16, tid);
      if (((n16 + TPB - 1) / TPB) == ITERS_FULL)
        asm volatile("s_wait_asynccnt 6" ::: "memory");  // chunk c done, c+1 in flight
      else
        asm volatile("s_wait_asynccnt 1" ::: "memory");
    } else {
      asm volatile("s_wait_asynccnt 0" ::: "memory");
    }
    __syncthreads();

    int p = c * CHUNK + tid;
    if (p < NUM_P) {
      const float* crow = confbuf + (c & 1) * CHUNK_FLOATS + tid * NUM_C;
      float ov = bto[p];
      int ti = bti[p];
      int cls = (ov < 0.5f) ? 0 : (labels_s[ti] + 1);
      if (cls > 0) {
        ++myNumPos;
        float4 t = ((const float4*)truths_s)[ti];
        float4 pr = ((const float4*)priors)[p];
        float px1 = pr.x - 0.5f * pr.z, py1 = pr.y - 0.5f * pr.w;
        float px2 = pr.x + 0.5f * pr.z, py2 = pr.y + 0.5f * pr.w;
        // faithful to source: encode against point-form prior
        float gcx = ((t.x + t.z) * 0.5f - px1) / (0.1f * px2);
        float gcy = ((t.y + t.w) * 0.5f - py1) / (0.1f * py2);
        float gw  = logf((t.z - t.x) / px2) * 5.0f;   // / 0.2
        float gh  = logf((t.w - t.y) / py2) * 5.0f;
        float4 ld = ((const float4*)loc)[(size_t)b * NUM_P + p];
        myLossL += smoothL1(ld.x - gcx) + smoothL1(ld.y - gcy) +
                   smoothL1(ld.z - gw) + smoothL1(ld.w - gh);
      }
      // per-row logsumexp (stride-21 LDS reads: conflict-free, 21 coprime 64)
      float m = crow[0];
#pragma unroll
      for (int j = 1; j < NUM_C; ++j) m = fmaxf(m, crow[j]);
      float s = 0.0f;
#pragma unroll
      for (int j = 0; j < NUM_C; ++j) s += expf(crow[j] - m);
      float lse = logf(s) + m;
      float cev = lse - crow[cls];
      ceArr[p] = cev;
      float keyf = (cls > 0) ? 0.0f : cev;         // positives zeroed for mining
      unsigned kb = __float_as_uint(keyf);
      kb = (kb & 0x80000000u) ? ~kb : (kb | 0x80000000u);  // order-preserving
      ukey[p] = kb;
    }
    __syncthreads();
  }

  // ---------------- Phase C: num_pos / num_neg ----------------
  red_i[tid] = myNumPos;
  __syncthreads();
  for (int s2 = TPB >> 1; s2 > 0; s2 >>= 1) {
    if (tid < s2) red_i[tid] += red_i[tid + s2];
    __syncthreads();
  }
  if (tid == 0) {
    int np = red_i[0];
    s_numpos = np;
    int nn = 3 * np; if (nn > NUM_P - 1) nn = NUM_P - 1;
    s_numneg = nn;
  }
  __syncthreads();
  const int numNeg = s_numneg;

  // ---------------- Phase D: radix-select k-th largest key ----------------
  unsigned T = 0u; int krem = 0;
  if (numNeg > 0) {                 // uniform branch: barriers are safe
    unsigned prefix = 0u, pmask = 0u;
    int k = numNeg;
    for (int shift = 24; shift >= 0; shift -= 8) {
      hist[tid] = 0u;
      __syncthreads();
      for (int p = tid; p < NUM_P; p += TPB) {
        unsigned uv = ukey[p];
        if ((uv & pmask) == prefix) atomicAdd(&hist[(uv >> shift) & 255u], 1u);
      }
      __syncthreads();
      if (tid == 0) {
        int cum = 0, sel = 0, kk = k;
        for (int bin = 255; bin >= 0; --bin) {
          cum += (int)hist[bin];
          if (cum >= kk) { sel = bin; kk -= cum - (int)hist[bin]; break; }
        }
        s_sel = sel; s_krem = kk;
      }
      __syncthreads();
      prefix |= ((unsigned)s_sel) << shift;
      pmask  |= (255u << shift);
      k = s_krem;
      __syncthreads();
    }
    T = prefix; krem = k;           // krem = #elements == T taken (lowest indices)
  }

  // ------- Phase E: stable tie-break (scan of equals) + loss_c sum -------
  const int per = (NUM_P + TPB - 1) / TPB;    // 35
  int lo = tid * per;
  int hi = lo + per; if (hi > NUM_P) hi = NUM_P;
  int eqc = 0;
  if (numNeg > 0)
    for (int p = lo; p < hi; ++p) eqc += (ukey[p] == T) ? 1 : 0;
  scan_s[tid] = eqc;
  __syncthreads();
  if (tid == 0) {
    int accu = 0;
    for (int i = 0; i < TPB; ++i) { int v = scan_s[i]; scan_s[i] = accu; accu += v; }
  }
  __syncthreads();
  int eqr = scan_s[tid];

  float myLossC = 0.0f;
  for (int p = lo; p < hi; ++p) {
    bool posP = (bto[p] >= 0.5f);
    bool negP = false;
    if (numNeg > 0) {
      unsigned uv = ukey[p];
      if (uv > T) negP = true;
      else if (uv == T) { negP = (eqr < krem); ++eqr; }
    }
    if (posP || negP) myLossC += ceArr[p];
  }

  // ---------------- Phase F: block reductions + global atomics ----------------
  red_f[tid] = myLossL;
  __syncthreads();
  for (int s2 = TPB >> 1; s2 > 0; s2 >>= 1) {
    if (tid < s2) red_f[tid] += red_f[tid + s2];
    __syncthreads();
  }
  if (tid == 0) atomicAdd(&acc[0], red_f[0]);
  __syncthreads();
  red_f[tid] = myLossC;
  __syncthreads();
  for (int s2 = TPB >> 1; s2 > 0; s2 >>= 1) {
    if (tid < s2) red_f[tid] += red_f[tid + s2];
    __syncthreads();
  }
  if (tid == 0) {
    atomicAdd(&acc[1], red_f[0]);
    atomicAdd(&acc[2], (float)s_numpos);
  }
}

__global__ void initAcc(float* acc) {
  if (threadIdx.x < 3) acc[threadIdx.x] = 0.0f;
}

__global__ void finalizeLoss(const float* acc, float* out) {
  float N = acc[2];
  out[0] = acc[0] / N;
  out[1] = acc[1] / N;
}

extern "C" void kernel_launch(void* const* d_in, const int* in_sizes, int n_in,
                              void* d_out, int out_size, void* d_ws, size_t ws_size,
                              hipStream_t stream) {
  const float* loc    = (const float*)d_in[0];
  const float* conf   = (const float*)d_in[1];
  const float* priors = (const float*)d_in[2];
  const float* truths = (const float*)d_in[3];
  const int*   labels = (const int*)d_in[4];
  float* out = (float*)d_out;
  float* acc = (float*)d_ws;

  int Bn = in_sizes[0] / (NUM_P * 4);

  size_t shmem = (size_t)(2 * CHUNK_FLOATS) * 4   // conf double buffer
               + (size_t)NUM_P * 4 * 3            // bto, ce, ukey
               + (size_t)NUM_O * 4 * 4            // truths_s
               + (size_t)NUM_O * 4 * 2            // labels_s, bpi
               + (size_t)NUM_P;                   // bti (bytes)
  (void)hipFuncSetAttribute((const void*)multibox_row_kernel,
                            hipFuncAttributeMaxDynamicSharedMemorySize, (int)shmem);

  initAcc<<<1, 32, 0, stream>>>(acc);
  multibox_row_kernel<<<Bn, TPB, shmem, stream>>>(loc, conf, priors, truths, labels, acc);
  finalizeLoss<<<1, 1, 0, stream>>>(acc, out);
}